// SkeletonLoss_9723805958291
// MI455X (gfx1250) — compile-verified
//
#include <hip/hip_runtime.h>
#include <stdint.h>

// ---------------------------------------------------------------------------
// Zhang-Suen skeleton loss for MI455X (gfx1250, wave32).
// Images are bitpacked: 512x512 bits -> 512 rows x 16 u32 words = 32KB/image.
// In-LDS working layout uses a row stride of 17 dwords so that a wave's 32
// consecutive rows hit 32 distinct LDS banks (17 coprime with 64) ->
// conflict-free neighbor reads in the hot Zhang-Suen loop.
// Kernel S: one block per (image, gt|pr): ballot-binarize -> iterate ZS in LDS
//           -> compact -> TDM tensor_store_from_lds (+ cooperative store).
// Kernel L: one block per image: TDM tensor_load_to_lds of both skeletons
//           (+ cooperative load), masked BCE, wave32 reduction -> per[b].
// Kernel F: total = sum(per)/8 -> d_out[0].
// ---------------------------------------------------------------------------

#define ZS_W 512
#define ZS_H 512
#define ZS_WORDS_PER_ROW 16            // 512 / 32
#define ZS_LDS_STRIDE    17            // padded row stride (dwords), bank-safe
#define ZS_WORDS_PER_IMG 8192          // 512 * 16 (compact)
#define ZS_PIX_PER_IMG   262144        // 512*512
#define ZS_IMG_BYTES     32768         // 8192 * 4

// ------------------------------- TDM helpers -------------------------------
#if __has_builtin(__builtin_amdgcn_tensor_load_to_lds) && \
    __has_builtin(__builtin_amdgcn_tensor_store_from_lds)
#define ZS_HAS_TDM 1
#if __clang_major__ >= 23
#warning "ZS: TDM builtins present, using 6-arg form (clang >= 23)"
#else
#warning "ZS: TDM builtins present, using 5-arg form (clang <= 22)"
#endif
typedef unsigned int zs_v4u __attribute__((ext_vector_type(4)));
typedef int          zs_v8i __attribute__((ext_vector_type(8)));
typedef int          zs_v4i __attribute__((ext_vector_type(4)));

// Build a D# for a 1-D tile of 8192 dwords (32KB) at (gaddr) <-> LDS byte
// offset lds_byte.  Layout per CDNA5 ISA.
__device__ __forceinline__ void zs_tdm_desc(unsigned long long gaddr,
                                            unsigned lds_byte,
                                            zs_v4u &g0, zs_v8i &g1) {
  g0[0] = 1u;                                        // count=1, user mode
  g0[1] = lds_byte;                                  // lds_addr[31:0]
  g0[2] = (unsigned)(gaddr & 0xFFFFFFFFull);         // global_addr[31:0]
  g0[3] = (unsigned)((gaddr >> 32) & 0x01FFFFFFull)  // global_addr[56:32]
          | 0x80000000u;                             // type = 2 ("image")
  g1[0] = 0x00020000;          // wg_mask=0, data_size=2 (4 bytes), no flags
  g1[1] = (int)(0x2000u << 16);// tensor_dim0 = 8192 (low 16 bits -> [31:16])
  g1[2] = 0x00010000;          // tensor_dim0 hi = 0, tensor_dim1 = 1
  g1[3] = (int)(0x2000u << 16);// tensor_dim1 hi = 0, tile_dim0 = 8192
  g1[4] = 0;                   // tile_dim1 = 0 (unused), tile_dim2 = 0
  g1[5] = 0x2000;              // tensor_dim0_stride = 8192
  g1[6] = 0;                   // stride hi = 0, tensor_dim1_stride lo = 0
  g1[7] = 0;                   // tensor_dim1_stride hi = 0
}

__device__ __forceinline__ void zs_tdm_load(unsigned long long gaddr,
                                            unsigned lds_byte) {
  zs_v4u g0; zs_v8i g1;
  zs_tdm_desc(gaddr, lds_byte, g0, g1);
  zs_v4i z4 = {0, 0, 0, 0};
#if __clang_major__ >= 23
  zs_v8i z8 = {0, 0, 0, 0, 0, 0, 0, 0};
  __builtin_amdgcn_tensor_load_to_lds(g0, g1, z4, z4, z8, 0);
#else
  __builtin_amdgcn_tensor_load_to_lds(g0, g1, z4, z4, 0);
#endif
}

__device__ __forceinline__ void zs_tdm_store(unsigned long long gaddr,
                                             unsigned lds_byte) {
  zs_v4u g0; zs_v8i g1;
  zs_tdm_desc(gaddr, lds_byte, g0, g1);
  zs_v4i z4 = {0, 0, 0, 0};
#if __clang_major__ >= 23
  zs_v8i z8 = {0, 0, 0, 0, 0, 0, 0, 0};
  __builtin_amdgcn_tensor_store_from_lds(g0, g1, z4, z4, z8, 0);
#else
  __builtin_amdgcn_tensor_store_from_lds(g0, g1, z4, z4, 0);
#endif
}
#else
#define ZS_HAS_TDM 0
#warning "ZS: TDM builtins NOT available on this toolchain - cooperative path only"
#endif

#if __has_builtin(__builtin_amdgcn_s_wait_tensorcnt)
#define ZS_WAIT_TENSOR() __builtin_amdgcn_s_wait_tensorcnt(0)
#else
#define ZS_WAIT_TENSOR() ((void)0)
#endif

// --------------------- bit-sliced Zhang-Suen primitives --------------------
// Add a 1-bit-per-pixel bitmap x into a bit-sliced 4-bit counter (b3,b2,b1,b0).
__device__ __forceinline__ void zs_add_bitplane(unsigned x, unsigned &b0,
                                                unsigned &b1, unsigned &b2,
                                                unsigned &b3) {
  unsigned c = b0 & x; b0 ^= x; x = c;
  c = b1 & x; b1 ^= x; x = c;
  c = b2 & x; b2 ^= x;
  b3 |= c;
}

// Track "exactly one" of the 0->1 transition bitmaps.
__device__ __forceinline__ void zs_trans(unsigned x, unsigned y,
                                         unsigned &any, unsigned &multi) {
  unsigned t = (~x) & y;
  multi |= any & t;
  any |= t;
}

// One ZS sub-iteration for row r (bit b of word w == pixel column 32*w+b).
// Reads the pre-substep image (stride ZS_LDS_STRIDE) from LDS, writes the new
// row into nrow[16].  Returns true if any pixel of this row was removed.
__device__ __forceinline__ bool zs_row(const unsigned *__restrict__ img,
                                       unsigned *__restrict__ nrow,
                                       int r, int step) {
  bool changed = false;
  const bool hasU = (r > 0);
  const bool hasD = (r < ZS_H - 1);
  const int baseU = (r - 1) * ZS_LDS_STRIDE;
  const int baseC = r * ZS_LDS_STRIDE;
  const int baseD = (r + 1) * ZS_LDS_STRIDE;

  unsigned aP = 0u, bP = 0u, cP = 0u;              // word w-1 of rows r-1,r,r+1
  unsigned aC = hasU ? img[baseU] : 0u;
  unsigned bC = img[baseC];
  unsigned cC = hasD ? img[baseD] : 0u;

  for (int w = 0; w < ZS_WORDS_PER_ROW; ++w) {
    const bool hasR = (w < ZS_WORDS_PER_ROW - 1);
    unsigned aN = (hasR && hasU) ? img[baseU + w + 1] : 0u;
    unsigned bN = hasR ? img[baseC + w + 1] : 0u;
    unsigned cN = (hasR && hasD) ? img[baseD + w + 1] : 0u;

    // Neighbor bitmaps aligned to the center pixel's bit position.
    unsigned P2 = aC;                              // N
    unsigned P6 = cC;                              // S
    unsigned P3 = (aC >> 1) | (aN << 31);          // NE (col+1)
    unsigned P9 = (aC << 1) | (aP >> 31);          // NW (col-1)
    unsigned P4 = (bC >> 1) | (bN << 31);          // E
    unsigned P8 = (bC << 1) | (bP >> 31);          // W
    unsigned P5 = (cC >> 1) | (cN << 31);          // SE
    unsigned P7 = (cC << 1) | (cP >> 31);          // SW

    // B = popcount of the 8 neighbors (bit-sliced carry-save counter).
    unsigned b0 = 0u, b1 = 0u, b2 = 0u, b3 = 0u;
    zs_add_bitplane(P2, b0, b1, b2, b3);
    zs_add_bitplane(P3, b0, b1, b2, b3);
    zs_add_bitplane(P4, b0, b1, b2, b3);
    zs_add_bitplane(P5, b0, b1, b2, b3);
    zs_add_bitplane(P6, b0, b1, b2, b3);
    zs_add_bitplane(P7, b0, b1, b2, b3);
    zs_add_bitplane(P8, b0, b1, b2, b3);
    zs_add_bitplane(P9, b0, b1, b2, b3);
    unsigned ge2 = b1 | b2 | b3;                   // B >= 2
    unsigned le6 = ~((b0 & b1 & b2) | b3);         // B <= 6 (not 7, not 8)

    // A == 1: exactly one 0->1 transition around P2,P3,...,P9,P2.
    unsigned any = 0u, multi = 0u;
    zs_trans(P2, P3, any, multi);
    zs_trans(P3, P4, any, multi);
    zs_trans(P4, P5, any, multi);
    zs_trans(P5, P6, any, multi);
    zs_trans(P6, P7, any, multi);
    zs_trans(P7, P8, any, multi);
    zs_trans(P8, P9, any, multi);
    zs_trans(P9, P2, any, multi);
    unsigned aEq1 = any & ~multi;

    unsigned c1, c2;
    if (step == 0) { c1 = ~(P2 & P4 & P6); c2 = ~(P4 & P6 & P8); }
    else           { c1 = ~(P2 & P4 & P8); c2 = ~(P2 & P6 & P8); }

    unsigned rem = bC & ge2 & le6 & aEq1 & c1 & c2;
    nrow[w] = bC & ~rem;
    changed |= (rem != 0u);

    aP = aC; aC = aN;
    bP = bC; bC = bN;
    cP = cC; cC = cN;
  }
  return changed;
}

// ------------------------------- Kernel S ----------------------------------
// blockIdx.x in [0,16): b = idx>>1, which = idx&1 (0 -> gt from target,
// 1 -> pr from pred; both thresholds are 0.5).  512 threads: thread == row.
__global__ __launch_bounds__(512) void zs_skeleton_kernel(
    const float *__restrict__ pred, const float *__restrict__ target,
    unsigned *__restrict__ skel) {
  __shared__ unsigned shm[ZS_H * ZS_LDS_STRIDE + 8]; // padded image + flag
  unsigned *img = shm;
  unsigned *flag = &shm[ZS_H * ZS_LDS_STRIDE];

  const int tid = threadIdx.x;
  const int b = blockIdx.x >> 1;
  const int which = blockIdx.x & 1;
  const float *src = (which ? pred : target) + (size_t)b * ZS_PIX_PER_IMG;

  const int lane = tid & 31;
  const int wave = tid >> 5;                        // 16 waves of 32

  // Binarize: each wave ballots 32 consecutive pixels into one packed word.
  // Flat word gw = it*16 + wave -> row it, word-in-row wave -> padded index.
  for (int it = 0; it < ZS_H; ++it) {
    int gw = it * ZS_WORDS_PER_ROW + wave;          // flat compact word
    float v = src[gw * 32 + lane];
    unsigned m = (unsigned)__ballot((int)(v > 0.5f));
    if (lane == 0) img[it * ZS_LDS_STRIDE + wave] = m;
  }
  if (tid == 0) *flag = 0u;
  __syncthreads();

  const int r = tid;                                // one row per thread
  unsigned nrow[ZS_WORDS_PER_ROW];

  for (;;) {
    bool ch0 = zs_row(img, nrow, r, 0);
    __syncthreads();                                // all reads of old image done
    for (int w = 0; w < ZS_WORDS_PER_ROW; ++w)
      img[r * ZS_LDS_STRIDE + w] = nrow[w];
    __syncthreads();                                // substep-0 image complete

    bool ch1 = zs_row(img, nrow, r, 1);
    __syncthreads();
    for (int w = 0; w < ZS_WORDS_PER_ROW; ++w)
      img[r * ZS_LDS_STRIDE + w] = nrow[w];
    if (ch0 | ch1) atomicOr(flag, 1u);
    __syncthreads();                                // image + flags complete

    unsigned f = *flag;
    __syncthreads();                                // everyone has read flag
    if (tid == 0) *flag = 0u;
    if (!f) break;                                  // uniform across block
    __syncthreads();                                // reset ordered before next OR
  }

  // Compact padded (stride 17) -> dense (stride 16) in place.  All reads are
  // staged to registers before the barrier, so overlap is race-free.
  unsigned tmp[ZS_WORDS_PER_ROW];
  for (int w = 0; w < ZS_WORDS_PER_ROW; ++w)
    tmp[w] = img[r * ZS_LDS_STRIDE + w];
  __syncthreads();
  for (int w = 0; w < ZS_WORDS_PER_ROW; ++w)
    img[(r << 4) + w] = tmp[w];
  __syncthreads();

  // Store converged skeleton: TDM DMA (one shot) + cooperative store of the
  // identical data (idempotent; keeps correctness independent of D# details).
  unsigned *dst = skel + (size_t)(which * 8 + b) * ZS_WORDS_PER_IMG;
#if ZS_HAS_TDM
  if (tid == 0) {
    zs_tdm_store((unsigned long long)(uintptr_t)dst, 0u);
    ZS_WAIT_TENSOR();
  }
#endif
  for (int w = 0; w < ZS_WORDS_PER_ROW; ++w)
    dst[(r << 4) + w] = tmp[w];
}

// ------------------------------- Kernel L ----------------------------------
// One block per image.  sk[0..8191] = gt skeleton, sk[8192..16383] = pr
// skeleton.  Binary maps are recomputed on the fly from pred/target.
__global__ __launch_bounds__(256) void zs_loss_kernel(
    const float *__restrict__ pred, const float *__restrict__ target,
    const unsigned *__restrict__ skel, float *__restrict__ per) {
  __shared__ unsigned sk[2 * ZS_WORDS_PER_IMG];     // 64KB
  const int tid = threadIdx.x;
  const int b = blockIdx.x;
  const unsigned *gsk = skel + (size_t)b * ZS_WORDS_PER_IMG;        // which=0
  const unsigned *psk = skel + (size_t)(8 + b) * ZS_WORDS_PER_IMG;  // which=1

#if ZS_HAS_TDM
  if (tid == 0) {
    zs_tdm_load((unsigned long long)(uintptr_t)gsk, 0u);
    zs_tdm_load((unsigned long long)(uintptr_t)psk, (unsigned)ZS_IMG_BYTES);
    ZS_WAIT_TENSOR();
  }
#endif
  // Cooperative load of the same data (identical values; benign overlap).
  for (int i = tid; i < ZS_WORDS_PER_IMG; i += 256) {
    sk[i] = gsk[i];
    sk[ZS_WORDS_PER_IMG + i] = psk[i];
  }
  __syncthreads();

  const float *p = pred + (size_t)b * ZS_PIX_PER_IMG;
  const float *t = target + (size_t)b * ZS_PIX_PER_IMG;

  float sum = 0.0f;
  unsigned cnt = 0u;
  for (int i = tid; i < ZS_PIX_PER_IMG; i += 256) {
    int word = i >> 5, bit = i & 31;
    unsigned g = (sk[word] >> bit) & 1u;                       // gt skeleton
    unsigned q = (sk[ZS_WORDS_PER_IMG + word] >> bit) & 1u;    // pr skeleton
    if (g | q) {
      float pv = p[i], tv = t[i];
      unsigned prb = (pv > 0.5f) ? 1u : 0u;                    // FILTER_P = 0.5
      unsigned gtb = (tv > 0.5f) ? 1u : 0u;
      unsigned sel = (g & (prb ^ 1u)) | (q & (gtb ^ 1u));
      if (sel) {
        float lp  = fmaxf(logf(pv), -100.0f);                  // LOG_CLAMP
        float l1p = fmaxf(log1pf(-pv), -100.0f);
        sum += -(tv * lp + (1.0f - tv) * l1p);
        cnt += 1u;
      }
    }
  }

  // wave32 shuffle reduction, then cross-wave via reused LDS.
  for (int o = 16; o > 0; o >>= 1) {
    sum += __shfl_down(sum, o, 32);
    cnt += __shfl_down(cnt, o, 32);
  }
  __syncthreads();                                  // done reading sk
  float *fs = reinterpret_cast<float *>(sk);        // [0..7] wave sums
  unsigned *us = sk + 8;                            // [8..15] wave counts
  if ((tid & 31) == 0) { fs[tid >> 5] = sum; us[tid >> 5] = cnt; }
  __syncthreads();
  if (tid == 0) {
    float S = 0.0f;
    unsigned C = 0u;
    for (int i = 0; i < 8; ++i) { S += fs[i]; C += us[i]; }
    per[b] = (C > 0u) ? (S / (float)C) : 0.0f;
  }
}

// ------------------------------- Kernel F ----------------------------------
__global__ void zs_final_kernel(const float *__restrict__ per,
                                float *__restrict__ out) {
  if (threadIdx.x == 0) {
    float s = 0.0f;
    for (int i = 0; i < 8; ++i) s += per[i];
    out[0] = s * 0.125f;                            // / batch (8)
  }
}

// ------------------------------- launcher ----------------------------------
extern "C" void kernel_launch(void *const *d_in, const int *in_sizes, int n_in,
                              void *d_out, int out_size, void *d_ws,
                              size_t ws_size, hipStream_t stream) {
  (void)in_sizes; (void)n_in; (void)out_size; (void)ws_size;
  const float *pred = (const float *)d_in[0];    // [8,1,512,512] f32
  const float *target = (const float *)d_in[1];  // [8,1,512,512] f32
  float *out = (float *)d_out;                   // [1] f32

  // Workspace layout: 16 packed skeletons (512KB) then per-image losses (32B).
  unsigned *skel = (unsigned *)d_ws;
  float *per = (float *)((char *)d_ws + (size_t)16 * ZS_IMG_BYTES);

  zs_skeleton_kernel<<<16, 512, 0, stream>>>(pred, target, skel);
  zs_loss_kernel<<<8, 256, 0, stream>>>(pred, target, skel, per);
  zs_final_kernel<<<1, 32, 0, stream>>>(per, out);
}